// GCN_24610162606454
// MI455X (gfx1250) — compile-verified
//
#include <hip/hip_runtime.h>

typedef float v2f __attribute__((ext_vector_type(2)));
typedef float v8f __attribute__((ext_vector_type(8)));

#define N_NODES 100000
#define N_EDGES 1600000

// ---------------- degree / normalization ----------------

__global__ void zero_u32_kernel(unsigned int* p, int n) {
    int i = blockIdx.x * blockDim.x + threadIdx.x;
    if (i < n) p[i] = 0u;
}

__global__ void deg_count_kernel(const int* __restrict__ dst,
                                 unsigned int* __restrict__ cnt, int E) {
    int e = blockIdx.x * blockDim.x + threadIdx.x;
    if (e < E) atomicAdd(&cnt[dst[e]], 1u);
}

__global__ void make_dis_kernel(const unsigned int* __restrict__ cnt,
                                float* __restrict__ dis, int n) {
    int i = blockIdx.x * blockDim.x + threadIdx.x;
    if (i < n) dis[i] = rsqrtf((float)cnt[i] + 1.0f);
}

// ---------------- WMMA GEMM: H[nrows x NOUT] = X[nrows x K] @ W[K x NOUT] ----------------
// One wave computes a 16 x NOUT tile using V_WMMA_F32_16X16X4_F32.
// A layout (16x4 f32): lanes 0-15 row r hold K=k0,k0+1 ; lanes 16-31 row r hold K=k0+2,k0+3
// C/D layout: VGPR v -> row = v + 8*(lane>=16), col = lane&15.

template<int K, int NOUT>
__global__ void __launch_bounds__(256)
gemm_wmma_kernel(const float* __restrict__ X, const float* __restrict__ W,
                 float* __restrict__ H, int nrows) {
    __shared__ float sW[K * NOUT];
    const int tid = threadIdx.x;
    for (int i = tid; i < K * NOUT; i += 256) sW[i] = W[i];
    __syncthreads();

    const int wave = tid >> 5;
    const int lane = tid & 31;
    const int half = lane >> 4;     // 0: K pair (k0,k0+1), 1: (k0+2,k0+3)
    const int r    = lane & 15;     // row within A tile / col within B,C tile

    const int row0 = (blockIdx.x * 8 + wave) * 16;
    if (row0 >= nrows) return;      // wave-uniform exit; EXEC stays all-ones for WMMA

    int arow = row0 + r;
    if (arow >= nrows) arow = nrows - 1;
    const float* __restrict__ xrow = X + (long long)arow * K;

    constexpr int NT = NOUT / 16;
    v8f acc[NT];
#pragma unroll
    for (int nt = 0; nt < NT; ++nt) {
        v8f z = {};
        acc[nt] = z;
    }

#pragma unroll 4
    for (int kt = 0; kt < K / 4; ++kt) {
        const int k0 = kt * 4 + 2 * half;
        v2f a;
        a.x = xrow[k0];
        a.y = xrow[k0 + 1];
#pragma unroll
        for (int nt = 0; nt < NT; ++nt) {
            v2f b;
            b.x = sW[(k0)     * NOUT + nt * 16 + r];
            b.y = sW[(k0 + 1) * NOUT + nt * 16 + r];
            acc[nt] = __builtin_amdgcn_wmma_f32_16x16x4_f32(
                false, a, false, b, (short)0, acc[nt], false, false);
        }
    }

#pragma unroll
    for (int nt = 0; nt < NT; ++nt) {
#pragma unroll
        for (int v = 0; v < 8; ++v) {
            const int row = row0 + v + 8 * half;
            const int col = nt * 16 + r;
            if (row < nrows) H[(long long)row * NOUT + col] = acc[nt][v];
        }
    }
}

// ---------------- self-loop init: acc = h * dis^2 ----------------

template<int F>
__global__ void self_init_kernel(const float* __restrict__ h,
                                 const float* __restrict__ dis,
                                 float* __restrict__ acc, int n) {
    long long idx = (long long)blockIdx.x * blockDim.x + threadIdx.x;
    if (idx < (long long)n * F) {
        int node = (int)(idx / F);
        float d = dis[node];
        acc[idx] = h[idx] * d * d;
    }
}

// ---------------- edge scatter: acc[dst] += h[src] * dis[src]*dis[dst] ----------------
// F contiguous threads per edge -> coalesced gather + coalesced f32 atomics.

template<int F>
__global__ void scatter_kernel(const int* __restrict__ src,
                               const int* __restrict__ dst,
                               const float* __restrict__ h,
                               const float* __restrict__ dis,
                               float* __restrict__ acc, int E) {
    long long idx = (long long)blockIdx.x * blockDim.x + threadIdx.x;
    if (idx >= (long long)E * F) return;
    const int e = (int)(idx >> __builtin_ctz(F));   // idx / F (F is power of two)
    const int f = (int)(idx & (F - 1));
    const int s = src[e];
    const int d = dst[e];
    const float coef = dis[s] * dis[d];
    atomicAdd(&acc[(long long)d * F + f], h[(long long)s * F + f] * coef);
}

// ---------------- epilogue: out = (acc + b) [relu] ----------------

template<int F, bool RELU>
__global__ void epilogue_kernel(const float* __restrict__ acc,
                                const float* __restrict__ b,
                                float* __restrict__ out, int n) {
    long long idx = (long long)blockIdx.x * blockDim.x + threadIdx.x;
    if (idx < (long long)n * F) {
        int f = (int)(idx & (F - 1));
        float v = acc[idx] + b[f];
        out[idx] = RELU ? fmaxf(v, 0.0f) : v;
    }
}

// ---------------- launch ----------------

static inline int cdiv(long long a, int b) { return (int)((a + b - 1) / b); }

extern "C" void kernel_launch(void* const* d_in, const int* in_sizes, int n_in,
                              void* d_out, int out_size, void* d_ws, size_t ws_size,
                              hipStream_t stream) {
    const float* x  = (const float*)d_in[0];
    const int*   ei = (const int*)d_in[1];      // [2, E] int32 (JAX default x64-off)
    const float* W1 = (const float*)d_in[2];
    const float* b1 = (const float*)d_in[3];
    const float* W2 = (const float*)d_in[4];
    const float* b2 = (const float*)d_in[5];
    const float* W3 = (const float*)d_in[6];
    const float* b3 = (const float*)d_in[7];
    float* out = (float*)d_out;

    const int* src = ei;            // row 0
    const int* dst = ei + N_EDGES;  // row 1

    char* ws = (char*)d_ws;
    unsigned int* cnt = (unsigned int*)ws;                    // 400000 B (pad to 400384)
    float* dis  = (float*)(ws + 400384);                      // 400000 B (pad to 800768)
    float* bufA = (float*)(ws + 800768);                      // 100000*64 f32 = 25.6 MB
    float* bufB = (float*)(ws + 800768 + 25600000);           // 100000*64 f32 = 25.6 MB

    const int N = N_NODES, E = N_EDGES;
    const int gemm_blocks = cdiv(cdiv(N, 16), 8);             // 8 waves/block, 16 rows/wave

    // normalization
    zero_u32_kernel<<<cdiv(N, 256), 256, 0, stream>>>(cnt, N);
    deg_count_kernel<<<cdiv(E, 256), 256, 0, stream>>>(dst, cnt, E);
    make_dis_kernel<<<cdiv(N, 256), 256, 0, stream>>>(cnt, dis, N);

    // ---- layer 1: x(128) -> 64, relu ----
    gemm_wmma_kernel<128, 64><<<gemm_blocks, 256, 0, stream>>>(x, W1, bufA, N);
    self_init_kernel<64><<<cdiv((long long)N * 64, 256), 256, 0, stream>>>(bufA, dis, bufB, N);
    scatter_kernel<64><<<cdiv((long long)E * 64, 256), 256, 0, stream>>>(src, dst, bufA, dis, bufB, E);
    epilogue_kernel<64, true><<<cdiv((long long)N * 64, 256), 256, 0, stream>>>(bufB, b1, bufA, N);

    // ---- layer 2: 64 -> 64, relu ----
    gemm_wmma_kernel<64, 64><<<gemm_blocks, 256, 0, stream>>>(bufA, W2, bufB, N);
    self_init_kernel<64><<<cdiv((long long)N * 64, 256), 256, 0, stream>>>(bufB, dis, bufA, N);
    scatter_kernel<64><<<cdiv((long long)E * 64, 256), 256, 0, stream>>>(src, dst, bufB, dis, bufA, E);
    epilogue_kernel<64, true><<<cdiv((long long)N * 64, 256), 256, 0, stream>>>(bufA, b2, bufB, N);

    // ---- layer 3: 64 -> 32, no relu; accumulate directly into d_out ----
    gemm_wmma_kernel<64, 32><<<gemm_blocks, 256, 0, stream>>>(bufB, W3, bufA, N);
    self_init_kernel<32><<<cdiv((long long)N * 32, 256), 256, 0, stream>>>(bufA, dis, out, N);
    scatter_kernel<32><<<cdiv((long long)E * 32, 256), 256, 0, stream>>>(src, dst, bufA, dis, out, E);
    epilogue_kernel<32, false><<<cdiv((long long)N * 32, 256), 256, 0, stream>>>(out, b3, out, N);
}